// Encoder_local_47004122087894
// MI455X (gfx1250) — compile-verified
//
#include <hip/hip_runtime.h>

typedef __attribute__((ext_vector_type(2))) float v2f;
typedef __attribute__((ext_vector_type(8))) float v8f;

#define D_IN  256
#define D_OUT 128
#define SA_STRIDE 260   // 256 + 4 pad: kills LDS bank conflicts, keeps 16B alignment

// ---------------------------------------------------------------------------
// Zero-fill helper (grid-stride)
// ---------------------------------------------------------------------------
__global__ __launch_bounds__(256) void k_zero(float* __restrict__ p, long n) {
    long i = (long)blockIdx.x * blockDim.x + threadIdx.x;
    long stride = (long)gridDim.x * blockDim.x;
    for (; i < n; i += stride) p[i] = 0.0f;
}

// ---------------------------------------------------------------------------
// h = L2normalize(relu(x @ W + b))  via V_WMMA_F32_16X16X4_F32
// Block = 256 threads (8 waves). Block owns a 16-row M tile; wave w owns
// N-columns [16w, 16w+16). K looped 0..256 step 4.
//
// Fragment layouts per CDNA5 ISA 7.12.2 (32-bit, wave32):
//   A 16x4:  lanes 0-15 -> M=lane, VGPR0=K0 VGPR1=K1; lanes 16-31 -> K2/K3
//   B 4x16:  lanes 0-15 -> N=lane, VGPR0=K0 VGPR1=K1; lanes 16-31 -> K2/K3
//   C 16x16: VGPR g -> row M = g + 8*(lane>=16), col N = lane&15
// ---------------------------------------------------------------------------
__global__ __launch_bounds__(256) void k_gemm(const float* __restrict__ x,
                                              const float* __restrict__ W,
                                              const float* __restrict__ bias,
                                              float* __restrict__ h) {
    __shared__ float sA[16 * SA_STRIDE];   // x tile 16x256 (padded)
    __shared__ float sH[16 * D_OUT];       // post-relu tile for norm reduction
    __shared__ float sInv[16];

    const int tid  = threadIdx.x;
    const int wave = tid >> 5;
    const int lane = tid & 31;
    const int m0   = blockIdx.x * 16;

    // Cooperative, coalesced load of the 16x256 x-tile as float4s.
    const float4* __restrict__ x4 = (const float4*)(x + (size_t)m0 * D_IN);
#pragma unroll
    for (int i = 0; i < 4; ++i) {
        int flat = tid + 256 * i;          // 0..1023 float4s
        int r  = flat >> 6;                // row 0..15
        int c4 = flat & 63;                // float4 col 0..63
        *(float4*)(sA + r * SA_STRIDE + c4 * 4) = x4[r * 64 + c4];
    }
    __syncthreads();

    const int n0    = wave * 16;
    const int laneN = lane & 15;           // A: M index; B: N index; C: N index
    const int hi    = lane >> 4;           // half-wave selects K pair / M+8

    v8f acc = {};
    for (int k = 0; k < D_IN; k += 4) {
        v2f a, bb;
        const int kk = k + 2 * hi;
        a.x  = sA[laneN * SA_STRIDE + kk];
        a.y  = sA[laneN * SA_STRIDE + kk + 1];
        bb.x = W[(size_t)kk       * D_OUT + n0 + laneN];
        bb.y = W[(size_t)(kk + 1) * D_OUT + n0 + laneN];
        acc = __builtin_amdgcn_wmma_f32_16x16x4_f32(
                  /*neg_a=*/false, a, /*neg_b=*/false, bb,
                  /*c_mod=*/(short)0, acc, /*reuse_a=*/false, /*reuse_b=*/false);
    }

    // bias + relu -> LDS tile
    const float bv = bias[n0 + laneN];
#pragma unroll
    for (int g = 0; g < 8; ++g) {
        float v = acc[g] + bv;
        v = v > 0.0f ? v : 0.0f;
        sH[(g + 8 * hi) * D_OUT + n0 + laneN] = v;
    }
    __syncthreads();

    // Row-wise L2 norm (16 rows; one thread each)
    if (tid < 16) {
        float s = 0.0f;
#pragma unroll 4
        for (int c = 0; c < D_OUT; ++c) {
            float v = sH[tid * D_OUT + c];
            s += v * v;
        }
        float nrm = sqrtf(s);
        sInv[tid] = 1.0f / fmaxf(nrm, 1e-12f);
    }
    __syncthreads();

    // Coalesced store of normalized tile
#pragma unroll
    for (int i = 0; i < 8; ++i) {
        int e = tid + 256 * i;             // 0..2047
        int r = e >> 7, c = e & (D_OUT - 1);
        h[(size_t)(m0 + r) * D_OUT + c] = sH[r * D_OUT + c] * sInv[r];
    }
}

// ---------------------------------------------------------------------------
// In-degree (float) via atomics
// ---------------------------------------------------------------------------
__global__ __launch_bounds__(256) void k_deg(const int* __restrict__ dst,
                                             float* __restrict__ deg, int E) {
    int e = blockIdx.x * 256 + threadIdx.x;
    if (e < E) atomicAdd(&deg[dst[e]], 1.0f);
}

// ---------------------------------------------------------------------------
// Edge scatter-add: one thread per (edge, feature)
// ---------------------------------------------------------------------------
__global__ __launch_bounds__(256) void k_scatter(const int* __restrict__ src,
                                                 const int* __restrict__ dst,
                                                 const float* __restrict__ feat,
                                                 float* __restrict__ outsum, int E) {
    long item = (long)blockIdx.x * 256 + threadIdx.x;
    if (item < (long)E * D_OUT) {
        int e = (int)(item >> 7);
        int c = (int)(item & (D_OUT - 1));
        atomicAdd(&outsum[(size_t)dst[e] * D_OUT + c],
                  feat[(size_t)src[e] * D_OUT + c]);
    }
}

// ---------------------------------------------------------------------------
// n1 = n1sum / max(deg, 1)   (in place)
// ---------------------------------------------------------------------------
__global__ __launch_bounds__(256) void k_div(float* __restrict__ n1,
                                             const float* __restrict__ deg, int Nn) {
    long i = (long)blockIdx.x * 256 + threadIdx.x;
    if (i < (long)Nn * D_OUT) {
        float d = fmaxf(deg[i >> 7], 1.0f);
        n1[i] = n1[i] / d;
    }
}

// ---------------------------------------------------------------------------
// out = L*n1 + (1-L)*(n2sum/denom)   (in place over n1)
// ---------------------------------------------------------------------------
__global__ __launch_bounds__(256) void k_combine(float* __restrict__ out2,
                                                 const float* __restrict__ n2sum,
                                                 const float* __restrict__ deg,
                                                 int Nn, float L) {
    long i = (long)blockIdx.x * 256 + threadIdx.x;
    if (i < (long)Nn * D_OUT) {
        float d = fmaxf(deg[i >> 7], 1.0f);
        out2[i] = L * out2[i] + (1.0f - L) * (n2sum[i] / d);
    }
}

// ---------------------------------------------------------------------------
extern "C" void kernel_launch(void* const* d_in, const int* in_sizes, int n_in,
                              void* d_out, int out_size, void* d_ws, size_t ws_size,
                              hipStream_t stream) {
    const float* x   = (const float*)d_in[0];
    const float* W   = (const float*)d_in[1];
    const float* b   = (const float*)d_in[2];
    const int*   src = (const int*)d_in[3];
    const int*   dst = (const int*)d_in[4];

    const int Nn = in_sizes[0] / D_IN;   // 50000 (multiple of 16)
    const int E  = in_sizes[3];          // 800000

    float* h  = (float*)d_out;                    // [Nn,128] first output
    float* n1 = h + (size_t)Nn * D_OUT;           // second half: neigh1 -> mean_h

    float* deg   = (float*)d_ws;                                  // [Nn]
    float* n2sum = deg + (((size_t)Nn + 255) & ~(size_t)255);     // [Nn,128]

    const long feats = (long)Nn * D_OUT;

    // zero accumulators (harness poisons buffers; we must init ourselves)
    k_zero<<<2048, 256, 0, stream>>>(n1, feats);
    k_zero<<<2048, 256, 0, stream>>>(deg, (long)Nn);
    k_zero<<<2048, 256, 0, stream>>>(n2sum, feats);

    // h = normalize(relu(x@W + b))  — WMMA f32 path
    k_gemm<<<Nn / 16, 256, 0, stream>>>(x, W, b, h);

    // in-degree
    k_deg<<<(E + 255) / 256, 256, 0, stream>>>(dst, deg, E);

    // neigh1 = segsum(h[src] -> dst) / denom
    const long items = (long)E * D_OUT;
    const int sgrid = (int)((items + 255) / 256);
    const int fgrid = (int)((feats + 255) / 256);
    k_scatter<<<sgrid, 256, 0, stream>>>(src, dst, h, n1, E);
    k_div<<<fgrid, 256, 0, stream>>>(n1, deg, Nn);

    // neigh2 = segsum(neigh1[src] -> dst) / denom ; mean_h = 0.7*n1 + 0.3*n2
    k_scatter<<<sgrid, 256, 0, stream>>>(src, dst, n1, n2sum, E);
    k_combine<<<fgrid, 256, 0, stream>>>(n1, n2sum, deg, Nn, 0.7f);
}